// GATEncoder_19980187861412
// MI455X (gfx1250) — compile-verified
//
#include <hip/hip_runtime.h>

#define N_NODES 10000
#define N_EDGES 160000
#define E_TOT   170000
#define BATCH   8192
#define NIN     128
#define H       256
#define HO2     512
#define DEC     1024

typedef float v2f __attribute__((ext_vector_type(2)));
typedef float v8f __attribute__((ext_vector_type(8)));

// ---- monotonic float <-> ordered-int encoding (for GLOBAL_ATOMIC_MAX_I32) ----
__device__ __forceinline__ int f2ord(float f) {
    int i = __float_as_int(f);
    return i >= 0 ? i : (int)(0x80000000u - (unsigned)i);
}
__device__ __forceinline__ float ord2f(int k) {
    return __int_as_float(k >= 0 ? k : (int)(0x80000000u - (unsigned)k));
}

// ---- edge_index may be int64 (as declared) or int32 (JAX x64 disabled) ----
__device__ __forceinline__ bool ei_is64(const void* ei) {
    const unsigned long long* p = (const unsigned long long*)ei;
    return ((p[0] | p[1] | p[2] | p[3]) >> 32) == 0ull;  // node ids < 10000
}
__device__ __forceinline__ int ei_get(const void* ei, bool is64, long long i) {
    return is64 ? (int)((const long long*)ei)[i] : ((const int*)ei)[i];
}

// ============================================================================
// GEMM: C[M,N] = epi(A[M,K] @ W[N,K]^T + bias)   via V_WMMA_F32_16X16X4_F32
// One wave computes a 16x64 tile (4 accumulators). Block = 4 waves -> 16x256.
// K-blocked by 16: stage 20 b64 loads, then 16 WMMAs, so loads from the next
// block overlap WMMAs of the current one (avoids s_wait_loadcnt 0 per WMMA).
// EPI: 0 = none, 1 = +bias, 2 = +bias then LeakyReLU(0.1)
// ============================================================================
template <int EPI>
__global__ void __launch_bounds__(128)
gemm_wmma(const float* __restrict__ A, const float* __restrict__ W,
          const float* __restrict__ bias, float* __restrict__ C,
          int M, int N, int K)
{
    const int wave = threadIdx.x >> 5;
    const int lane = threadIdx.x & 31;
    const int L    = lane & 15;     // row-in-tile (A) / col-in-tile (B,C)
    const int hi   = lane >> 4;     // K-half selector per ISA layout
    const int m0   = blockIdx.x << 4;
    const int n0   = ((blockIdx.y << 2) + wave) << 6;

    const float* Ap = A + (size_t)(m0 + L) * K + 2 * hi;
    const float* Wp = W + (size_t)(n0 + L) * K + 2 * hi;   // B[k][n] = W[n][k]
    const size_t WS = (size_t)16 * K;

    v8f acc0 = {0.f,0.f,0.f,0.f,0.f,0.f,0.f,0.f};
    v8f acc1 = acc0, acc2 = acc0, acc3 = acc0;

    for (int k = 0; k < K; k += 16) {        // K is always a multiple of 16
        v2f a[4], b0[4], b1[4], b2[4], b3[4];
#pragma unroll
        for (int u = 0; u < 4; ++u) {        // issue all 20 loads up front
            const int kk = k + 4 * u;
            a[u]  = *(const v2f*)(Ap + kk);
            b0[u] = *(const v2f*)(Wp + kk);
            b1[u] = *(const v2f*)(Wp + WS + kk);
            b2[u] = *(const v2f*)(Wp + 2 * WS + kk);
            b3[u] = *(const v2f*)(Wp + 3 * WS + kk);
        }
#pragma unroll
        for (int u = 0; u < 4; ++u) {        // then the 16 WMMAs
            acc0 = __builtin_amdgcn_wmma_f32_16x16x4_f32(false, a[u], false, b0[u], (short)0, acc0, false, false);
            acc1 = __builtin_amdgcn_wmma_f32_16x16x4_f32(false, a[u], false, b1[u], (short)0, acc1, false, false);
            acc2 = __builtin_amdgcn_wmma_f32_16x16x4_f32(false, a[u], false, b2[u], (short)0, acc2, false, false);
            acc3 = __builtin_amdgcn_wmma_f32_16x16x4_f32(false, a[u], false, b3[u], (short)0, acc3, false, false);
        }
    }

    float bz0 = 0.f, bz1 = 0.f, bz2 = 0.f, bz3 = 0.f;
    if (EPI >= 1) {
        bz0 = bias[n0 + L];      bz1 = bias[n0 + 16 + L];
        bz2 = bias[n0 + 32 + L]; bz3 = bias[n0 + 48 + L];
    }
    // C/D layout: VGPR j of lane(L,hi) holds C[m0 + 8*hi + j][n0 + 16*t + L]
    float* Cp = C + (size_t)(m0 + 8 * hi) * N + n0 + L;
#pragma unroll
    for (int j = 0; j < 8; ++j) {
        float v0 = acc0[j], v1 = acc1[j], v2 = acc2[j], v3 = acc3[j];
        if (EPI >= 1) { v0 += bz0; v1 += bz1; v2 += bz2; v3 += bz3; }
        if (EPI == 2) {
            v0 = v0 > 0.f ? v0 : 0.1f * v0;  v1 = v1 > 0.f ? v1 : 0.1f * v1;
            v2 = v2 > 0.f ? v2 : 0.1f * v2;  v3 = v3 > 0.f ? v3 : 0.1f * v3;
        }
        size_t r = (size_t)j * N;
        Cp[r] = v0; Cp[r + 16] = v1; Cp[r + 32] = v2; Cp[r + 48] = v3;
    }
}

// ============================================================================
// LayerNorm(256) + ReLU, in-place. One 256-thread block per row.
// ============================================================================
__global__ void __launch_bounds__(256)
ln_relu(float* __restrict__ h, const float* __restrict__ g, const float* __restrict__ b)
{
    const int row = blockIdx.x, t = threadIdx.x;
    float v = h[(size_t)row * H + t];
    __shared__ float s1[256], s2[256];
    s1[t] = v; s2[t] = v * v;
    __syncthreads();
    for (int off = 128; off; off >>= 1) {
        if (t < off) { s1[t] += s1[t + off]; s2[t] += s2[t + off]; }
        __syncthreads();
    }
    const float mu  = s1[0] * (1.0f / H);
    const float var = s2[0] * (1.0f / H) - mu * mu;
    const float r   = rsqrtf(var + 1e-5f);
    float y = (v - mu) * r * g[t] + b[t];
    h[(size_t)row * H + t] = fmaxf(y, 0.f);
}

// ============================================================================
// a_src[n][hd] = <xs[n,hd,:], att_src[hd,:]>, same for a_dst. 128 thr / node.
// ============================================================================
__global__ void __launch_bounds__(128)
att_scores(const float* __restrict__ xs, const float* __restrict__ as,
           const float* __restrict__ ad, float* __restrict__ a_src,
           float* __restrict__ a_dst)
{
    const int n = blockIdx.x, t = threadIdx.x;
    const float* xr = xs + (size_t)n * HO2;
    const float x0 = xr[t], x1 = xr[t + 128], x2 = xr[t + 256], x3 = xr[t + 384];
    __shared__ float sm[4][128];
    sm[0][t] = x0 * as[t]       + x1 * as[128 + t];   // src head 0
    sm[1][t] = x2 * as[256 + t] + x3 * as[384 + t];   // src head 1
    sm[2][t] = x0 * ad[t]       + x1 * ad[128 + t];   // dst head 0
    sm[3][t] = x2 * ad[256 + t] + x3 * ad[384 + t];   // dst head 1
    __syncthreads();
    for (int off = 64; off; off >>= 1) {
        if (t < off) {
#pragma unroll
            for (int q = 0; q < 4; ++q) sm[q][t] += sm[q][t + off];
        }
        __syncthreads();
    }
    if (t < 2)           a_src[(size_t)n * 2 + t]       = sm[t][0];
    else if (t < 4)      a_dst[(size_t)n * 2 + (t - 2)] = sm[t][0];
}

// ============================================================================
// Init: zero agg + z, set m to -inf (ordered-int INT_MIN)
// ============================================================================
__global__ void init_ws(float* __restrict__ agg, float* __restrict__ z, int* __restrict__ m)
{
    const int idx = blockIdx.x * blockDim.x + threadIdx.x;
    if (idx < BATCH * HO2) agg[idx] = 0.f;
    if (idx < BATCH * 2) { z[idx] = 0.f; m[idx] = (int)0x80000000; }
}

// ============================================================================
// Edge pass 1: e = leaky(a_src[src]+a_dst[tgt], 0.2); atomicMax into m[tgt]
// ============================================================================
__global__ void __launch_bounds__(256)
edge_pass1(const void* __restrict__ ei, const float* __restrict__ a_src,
           const float* __restrict__ a_dst, float* __restrict__ ew, int* __restrict__ m)
{
    const bool is64 = ei_is64(ei);
    const int i = blockIdx.x * blockDim.x + threadIdx.x;
    if (i >= E_TOT) return;
    int s, t;
    if (i < N_EDGES) { s = ei_get(ei, is64, i); t = ei_get(ei, is64, (long long)N_EDGES + i); }
    else             { s = t = i - N_EDGES; }
    if (t >= BATCH) return;   // targets beyond batch never reach the output
#pragma unroll
    for (int hd = 0; hd < 2; ++hd) {
        float e = a_src[(size_t)s * 2 + hd] + a_dst[(size_t)t * 2 + hd];
        e = e > 0.f ? e : 0.2f * e;
        ew[(size_t)i * 2 + hd] = e;
        atomicMax(&m[t * 2 + hd], f2ord(e));
    }
}

// ============================================================================
// Edge pass 2: w = exp(e - m[tgt]); atomicAdd into z[tgt]; store w
// ============================================================================
__global__ void __launch_bounds__(256)
edge_pass2(const void* __restrict__ ei, float* __restrict__ ew,
           const int* __restrict__ m, float* __restrict__ z)
{
    const bool is64 = ei_is64(ei);
    const int i = blockIdx.x * blockDim.x + threadIdx.x;
    if (i >= E_TOT) return;
    int t;
    if (i < N_EDGES) t = ei_get(ei, is64, (long long)N_EDGES + i);
    else             t = i - N_EDGES;
    if (t >= BATCH) return;
#pragma unroll
    for (int hd = 0; hd < 2; ++hd) {
        float w = expf(ew[(size_t)i * 2 + hd] - ord2f(m[t * 2 + hd]));
        ew[(size_t)i * 2 + hd] = w;
        atomicAdd(&z[t * 2 + hd], w);
    }
}

// ============================================================================
// Edge pass 3: agg[tgt] += xs[src] * (w / z[tgt]).  One 128-thr block / edge,
// each thread scatters 4 consecutive floats (float4 gather, 4 f32 atomics).
// ============================================================================
__global__ void __launch_bounds__(128)
edge_pass3(const void* __restrict__ ei, const float* __restrict__ xs,
           const float* __restrict__ ew, const float* __restrict__ z,
           float* __restrict__ agg)
{
    const bool is64 = ei_is64(ei);
    const int i = blockIdx.x;
    int s, t;
    if (i < N_EDGES) { s = ei_get(ei, is64, i); t = ei_get(ei, is64, (long long)N_EDGES + i); }
    else             { s = t = i - N_EDGES; }
    if (t >= BATCH) return;
    const int j0 = threadIdx.x * 4;        // 0..508
    const int hd = j0 >> 8;                // head 0 or 1
    const float alpha = ew[(size_t)i * 2 + hd] / z[t * 2 + hd];
    const float4 xv = *(const float4*)(xs + (size_t)s * HO2 + j0);
    float* ag = agg + (size_t)t * HO2 + j0;
    atomicAdd(ag + 0, xv.x * alpha);
    atomicAdd(ag + 1, xv.y * alpha);
    atomicAdd(ag + 2, xv.z * alpha);
    atomicAdd(ag + 3, xv.w * alpha);
}

// ============================================================================
// out = ELU_{0.1}(agg + gat_bias + skip), in-place into agg
// ============================================================================
__global__ void __launch_bounds__(256)
combine(float* __restrict__ agg, const float* __restrict__ skip,
        const float* __restrict__ gbias)
{
    const int idx = blockIdx.x * blockDim.x + threadIdx.x;   // exact: 8192*512
    const int col = idx & (HO2 - 1);
    float v = agg[idx] + gbias[col] + skip[idx];
    agg[idx] = v > 0.f ? v : 0.1f * (expf(v) - 1.0f);
}

// ============================================================================
// Decoder final: y[r] = <d[r,:], w2> + b2.  One wave32 per row.
// ============================================================================
__global__ void __launch_bounds__(256)
dec2_kernel(const float* __restrict__ d, const float* __restrict__ w2,
            const float* __restrict__ b2, float* __restrict__ y)
{
    const int wave = threadIdx.x >> 5, lane = threadIdx.x & 31;
    const int row = (blockIdx.x << 3) + wave;
    const float* dr = d + (size_t)row * DEC;
    float s = 0.f;
    for (int k = lane; k < DEC; k += 32) s += dr[k] * w2[k];
#pragma unroll
    for (int off = 16; off; off >>= 1) s += __shfl_down(s, off, 32);
    if (lane == 0) y[row] = s + b2[0];
}

// ============================================================================
extern "C" void kernel_launch(void* const* d_in, const int* in_sizes, int n_in,
                              void* d_out, int out_size, void* d_ws, size_t ws_size,
                              hipStream_t stream)
{
    const float* x       = (const float*)d_in[0];
    const void*  ei      = d_in[1];
    const float* enc_w1  = (const float*)d_in[3];
    const float* enc_b1  = (const float*)d_in[4];
    const float* ln_g    = (const float*)d_in[5];
    const float* ln_b    = (const float*)d_in[6];
    const float* enc_w2  = (const float*)d_in[7];
    const float* enc_b2  = (const float*)d_in[8];
    const float* gat_w   = (const float*)d_in[9];
    const float* att_src = (const float*)d_in[10];
    const float* att_dst = (const float*)d_in[11];
    const float* gat_b   = (const float*)d_in[12];
    const float* skip_w  = (const float*)d_in[13];
    const float* skip_b  = (const float*)d_in[14];
    const float* dec_w1  = (const float*)d_in[15];
    const float* dec_b1  = (const float*)d_in[16];
    const float* dec_w2  = (const float*)d_in[17];
    const float* dec_b2  = (const float*)d_in[18];
    float* y = (float*)d_out;

    // ---- workspace layout (floats); d-buffer reuses xs+skip region ----
    float* ws = (float*)d_ws;
    float* xs    = ws;                                        // [N_NODES, 512]
    float* skip  = xs   + (size_t)N_NODES * HO2;              // [BATCH, 512]
    float* agg   = skip + (size_t)BATCH * HO2;                // [BATCH, 512] (also "out")
    float* hp    = agg  + (size_t)BATCH * HO2;                // [N_NODES, 256] h_pre/h1
    float* h     = hp   + (size_t)N_NODES * H;                // [N_NODES, 256]
    float* asrc  = h    + (size_t)N_NODES * H;                // [N_NODES, 2]
    float* adst  = asrc + (size_t)N_NODES * 2;                // [N_NODES, 2]
    int*   mbuf  = (int*)(adst + (size_t)N_NODES * 2);        // [BATCH, 2]
    float* zbuf  = (float*)(mbuf + (size_t)BATCH * 2);        // [BATCH, 2]
    float* ewbuf = zbuf + (size_t)BATCH * 2;                  // [E_TOT, 2]
    float* dbuf  = ws;   // [BATCH, 1024] — xs & skip dead by decoder time

    // init softmax/agg state
    init_ws<<<(BATCH * HO2) / 256, 256, 0, stream>>>(agg, zbuf, mbuf);

    // encoder
    gemm_wmma<1><<<dim3(N_NODES / 16, 1), 128, 0, stream>>>(x, enc_w1, enc_b1, hp, N_NODES, H, NIN);
    ln_relu<<<N_NODES, 256, 0, stream>>>(hp, ln_g, ln_b);
    gemm_wmma<1><<<dim3(N_NODES / 16, 1), 128, 0, stream>>>(hp, enc_w2, enc_b2, h, N_NODES, H, H);

    // GAT linear + skip projection
    gemm_wmma<0><<<dim3(N_NODES / 16, 2), 128, 0, stream>>>(h, gat_w, nullptr, xs, N_NODES, HO2, H);
    gemm_wmma<1><<<dim3(BATCH / 16, 2), 128, 0, stream>>>(h, skip_w, skip_b, skip, BATCH, HO2, H);

    // attention
    att_scores<<<N_NODES, 128, 0, stream>>>(xs, att_src, att_dst, asrc, adst);
    edge_pass1<<<(E_TOT + 255) / 256, 256, 0, stream>>>(ei, asrc, adst, ewbuf, mbuf);
    edge_pass2<<<(E_TOT + 255) / 256, 256, 0, stream>>>(ei, ewbuf, mbuf, zbuf);
    edge_pass3<<<E_TOT, 128, 0, stream>>>(ei, xs, ewbuf, zbuf, agg);

    // combine + ELU
    combine<<<(BATCH * HO2) / 256, 256, 0, stream>>>(agg, skip, gat_b);

    // decoder
    gemm_wmma<2><<<dim3(BATCH / 16, 4), 128, 0, stream>>>(agg, dec_w1, dec_b1, dbuf, BATCH, DEC, HO2);
    dec2_kernel<<<BATCH / 8, 256, 0, stream>>>(dbuf, dec_w2, dec_b2, y);
}